// AccumulatorRNNDecision_40939628265828
// MI455X (gfx1250) — compile-verified
//
#include <hip/hip_runtime.h>
#include <math.h>
#include <stdint.h>

// Problem constants (match reference)
#define B_    16384
#define C_    4
#define H_    8
#define H2_   (H_ / 2)
#define T_    120
#define BLOCK 256
#define NW    (BLOCK / 32)      // waves per block
#define CHUNK 4                 // time-steps prefetched per TDM descriptor
#define NCHUNK (T_ / CHUNK)
#define TILE0 (32 * H_)         // 256 f32 elements per wave per step (1 KB)

static __device__ constexpr float DT_MS  = 10.0f;
static __device__ constexpr float THRESH = 0.5f;
static __device__ constexpr float ALPHA_ = 0.2f;

#if defined(__AMDGCN__)
#if __has_builtin(__builtin_amdgcn_tensor_load_to_lds) && __has_builtin(__builtin_amdgcn_s_wait_tensorcnt)
#define HAVE_TDM 1
#endif
#endif

typedef float        f32x2 __attribute__((ext_vector_type(2)));
typedef unsigned int u32x4 __attribute__((ext_vector_type(4)));
typedef int          i32x4 __attribute__((ext_vector_type(4)));
typedef int          i32x8 __attribute__((ext_vector_type(8)));

__device__ __forceinline__ float htanh(float x) {
#if defined(__AMDGCN__) && __has_builtin(__builtin_amdgcn_tanhf)
  return __builtin_amdgcn_tanhf(x);           // v_tanh_f32 (trans pipe)
#elif defined(__AMDGCN__) && __has_builtin(__builtin_amdgcn_tanh_f32)
  return __builtin_amdgcn_tanh_f32(x);
#else
  return tanhf(x);
#endif
}

__device__ __forceinline__ f32x2 pk_fma(f32x2 a, f32x2 b, f32x2 c) {
  return __builtin_elementwise_fma(a, b, c);  // v_pk_fma_f32
}

#ifdef HAVE_TDM
// One TDM descriptor: 2D tile = CHUNK rows x TILE0 f32 elements,
// row stride = B*C*H elements (one time step), into this wave's LDS slot.
__device__ __forceinline__ void tdm_load_chunk(const float* noise, int t0,
                                               uint32_t wave_elem_base,
                                               uint32_t lds_off) {
  uint64_t g = (uint64_t)(uintptr_t)noise +
               ((uint64_t)t0 * (uint64_t)(B_ * C_) + (uint64_t)wave_elem_base) *
                   (uint64_t)(H_ * 4);
  uint32_t glo = (uint32_t)__builtin_amdgcn_readfirstlane((int)(uint32_t)g);
  uint32_t ghi = (uint32_t)__builtin_amdgcn_readfirstlane((int)(uint32_t)(g >> 32));
  uint32_t lof = (uint32_t)__builtin_amdgcn_readfirstlane((int)lds_off);

  u32x4 g0;
  g0[0] = 1u;                                   // count=1, user descriptor
  g0[1] = lof;                                  // lds_addr (bytes)
  g0[2] = glo;                                  // global_addr[31:0]
  g0[3] = (ghi & 0x01FFFFFFu) | (2u << 30);     // global_addr[56:32] | type=2

  i32x8 g1;
  g1[0] = (int)(2u << 16);                      // data_size = 4 bytes; wg_mask=0
  g1[1] = (int)((uint32_t)TILE0 << 16);         // tensor_dim0[15:0] = 256
  g1[2] = (int)((uint32_t)CHUNK << 16);         // tensor_dim1[15:0] = CHUNK
  g1[3] = (int)((uint32_t)TILE0 << 16);         // tile_dim0 = 256
  g1[4] = (int)(uint32_t)CHUNK;                 // tile_dim1 = CHUNK, tile_dim2 = 0
  g1[5] = (int)(B_ * C_ * H_);                  // tensor_dim0_stride[31:0] = 524288
  g1[6] = 0;
  g1[7] = 0;

  i32x4 z4 = {0, 0, 0, 0};
#if __clang_major__ >= 23
  i32x8 z8 = {0, 0, 0, 0, 0, 0, 0, 0};
  __builtin_amdgcn_tensor_load_to_lds(g0, g1, z4, z4, z8, 0);
#else
  __builtin_amdgcn_tensor_load_to_lds(g0, g1, z4, z4, 0);
#endif
}
#endif  // HAVE_TDM

__global__ __launch_bounds__(BLOCK) void acc_rnn_kernel(
    const float* __restrict__ logits,        // [B,C]
    const float* __restrict__ input_scale,   // [1]
    const float* __restrict__ noise_std,     // [1]
    const float* __restrict__ input_proj_w,  // [H,1]
    const float* __restrict__ input_proj_b,  // [H]
    const float* __restrict__ self_proj_w,   // [H,H]
    const float* __restrict__ comp_proj_w,   // [H,1]
    const float* __restrict__ evidence_w,    // [1,H]
    const float* __restrict__ evidence_b,    // [1]
    const float* __restrict__ class_bias,    // [C,H]
    const float* __restrict__ competition,   // [C,C]
    const float* __restrict__ noise,         // [T,B,C,H]
    float* __restrict__ out)                 // [B,C]
{
#ifdef HAVE_TDM
  __shared__ __align__(16) float lbuf[2][NW][CHUNK][TILE0];  // 64 KiB
#endif
  const int tid = blockIdx.x * BLOCK + threadIdx.x;  // (b*C + c)
  const int c   = tid & (C_ - 1);
  const int wv  = threadIdx.x >> 5;
  const int ln  = threadIdx.x & 31;
  (void)wv; (void)ln;

  // Uniform scalars / weights -> scalar loads (SGPR-resident)
  const float iscale = input_scale[0];
  const float evb    = evidence_b[0];
  const f32x2 nstd2  = {noise_std[0], noise_std[0]};
  const f32x2 om2    = {1.0f - ALPHA_, 1.0f - ALPHA_};  // (1-alpha)

  // Scaled-state recurrence: state = alpha * u.
  //   u' = (1-alpha)*u + cand,  matvec uses W' = alpha*W,  ev uses evw' = alpha*evw.
  // Pack k-dimension in pairs for v_pk_fma_f32.
  f32x2 Wp[H_][H2_];       // Wp[h][k2] = alpha * {W[2k2][h], W[2k2+1][h]}  (uniform)
  f32x2 evw2[H2_], cpw2[H2_], inpc2[H2_];
#pragma unroll
  for (int k2 = 0; k2 < H2_; ++k2) {
    evw2[k2] = (f32x2){ALPHA_ * evidence_w[2 * k2], ALPHA_ * evidence_w[2 * k2 + 1]};
    cpw2[k2] = (f32x2){comp_proj_w[2 * k2], comp_proj_w[2 * k2 + 1]};
#pragma unroll
    for (int h = 0; h < H_; ++h)
      Wp[h][k2] = (f32x2){ALPHA_ * self_proj_w[(2 * k2) * H_ + h],
                          ALPHA_ * self_proj_w[(2 * k2 + 1) * H_ + h]};
  }
  float ccol[C_];  // column c of competition (per-lane)
#pragma unroll
  for (int j = 0; j < C_; ++j) ccol[j] = competition[j * C_ + c];

  // Input embed: constant over time, fold into per-thread registers.
  const float rl = fmaxf(logits[tid] * iscale, 0.0f);
#pragma unroll
  for (int k2 = 0; k2 < H2_; ++k2) {
    inpc2[k2].x = fmaf(rl, input_proj_w[2 * k2],
                       input_proj_b[2 * k2] + class_bias[c * H_ + 2 * k2]);
    inpc2[k2].y = fmaf(rl, input_proj_w[2 * k2 + 1],
                       input_proj_b[2 * k2 + 1] + class_bias[c * H_ + 2 * k2 + 1]);
  }

  f32x2 u2[H2_];
#pragma unroll
  for (int k2 = 0; k2 < H2_; ++k2) u2[k2] = (f32x2){0.0f, 0.0f};
  float ev = evb;   // evidence of current state (state0 = 0)
  int tfirst = T_;  // sentinel: never crossed

#ifdef HAVE_TDM
  const uint32_t wbase = (uint32_t)(blockIdx.x * BLOCK + wv * 32);
  uint32_t loff[2];
  loff[0] = (uint32_t)(uintptr_t)&lbuf[0][wv][0][0];  // generic ptr low 32 = LDS offset
  loff[1] = (uint32_t)(uintptr_t)&lbuf[1][wv][0][0];
  tdm_load_chunk(noise, 0, wbase, loff[0]);
#endif

  // One time step; reads this thread's 8 noise values from `np`.
  auto do_step = [&](const float4* np, int t) {
    float4 n0 = np[0], n1 = np[1];
    f32x2 nz2[H2_] = {{n0.x, n0.y}, {n0.z, n0.w}, {n1.x, n1.y}, {n1.z, n1.w}};

    // comp[b,c] = sum_j ev[b,j] * competition[j,c]  (quad broadcast over classes)
    float comp = 0.0f;
#pragma unroll
    for (int j = 0; j < C_; ++j)
      comp = fmaf(__shfl(ev, j, 4), ccol[j], comp);
    const f32x2 comp2 = {comp, comp};

    // Preactivation: inp + comp*cproj + noise*std + W'*u   (packed over k)
    f32x2 a2[H2_];
#pragma unroll
    for (int k2 = 0; k2 < H2_; ++k2) {
      a2[k2] = pk_fma(comp2, cpw2[k2], inpc2[k2]);
      a2[k2] = pk_fma(nz2[k2], nstd2, a2[k2]);
    }
#pragma unroll
    for (int h = 0; h < H_; ++h) {
      const float uh = (h & 1) ? u2[h >> 1].y : u2[h >> 1].x;
      const f32x2 us = {uh, uh};
#pragma unroll
      for (int k2 = 0; k2 < H2_; ++k2) a2[k2] = pk_fma(us, Wp[h][k2], a2[k2]);
    }

    // cand = tanh(a);  u' = (1-alpha)*u + cand;  ev' = sum alpha*evw*u' + evb
    f32x2 e2 = {evb, 0.0f};
#pragma unroll
    for (int k2 = 0; k2 < H2_; ++k2) {
      f32x2 cand2 = {htanh(a2[k2].x), htanh(a2[k2].y)};
      u2[k2] = pk_fma(u2[k2], om2, cand2);
      e2 = pk_fma(u2[k2], evw2[k2], e2);
    }
    ev = e2.x + e2.y;
    // Branch-free first-crossing: select + min.
    tfirst = min(tfirst, (ev > THRESH) ? t : T_);
  };

#ifdef HAVE_TDM
  // Steady-state: issue next chunk's TDM, wait for current, consume 4 steps.
  for (int ch = 0; ch < NCHUNK - 1; ++ch) {
    const int cur = ch & 1;
    tdm_load_chunk(noise, (ch + 1) * CHUNK, wbase, loff[cur ^ 1]);
    __builtin_amdgcn_s_wait_tensorcnt(1);  // current chunk complete
    asm volatile("" ::: "memory");         // TDM wrote LDS behind the compiler's back
    const float* nsrc = &lbuf[cur][wv][0][ln * H_];
#pragma unroll
    for (int s = 0; s < CHUNK; ++s)
      do_step((const float4*)(nsrc + s * TILE0), ch * CHUNK + s);
  }
  {  // Last chunk (no further prefetch).
    __builtin_amdgcn_s_wait_tensorcnt(0);
    asm volatile("" ::: "memory");
    const float* nsrc = &lbuf[(NCHUNK - 1) & 1][wv][0][ln * H_];
#pragma unroll
    for (int s = 0; s < CHUNK; ++s)
      do_step((const float4*)(nsrc + s * TILE0), (NCHUNK - 1) * CHUNK + s);
  }
#else
  for (int t = 0; t < T_; ++t)
    do_step((const float4*)(noise + ((size_t)t * (B_ * C_) + (size_t)tid) * H_), t);
#endif

  const float tdec = (tfirst < T_) ? (float)(tfirst + 1) * DT_MS : (float)T_ * DT_MS;
  out[tid] = tdec * 0.001f;
}

extern "C" void kernel_launch(void* const* d_in, const int* in_sizes, int n_in,
                              void* d_out, int out_size, void* d_ws, size_t ws_size,
                              hipStream_t stream) {
  (void)in_sizes; (void)n_in; (void)out_size; (void)d_ws; (void)ws_size;
  acc_rnn_kernel<<<(B_ * C_) / BLOCK, BLOCK, 0, stream>>>(
      (const float*)d_in[0],  (const float*)d_in[1],  (const float*)d_in[2],
      (const float*)d_in[3],  (const float*)d_in[4],  (const float*)d_in[5],
      (const float*)d_in[6],  (const float*)d_in[7],  (const float*)d_in[8],
      (const float*)d_in[9],  (const float*)d_in[10], (const float*)d_in[11],
      (float*)d_out);
}